// QCNNClassifier_59725815218534
// MI455X (gfx1250) — compile-verified
//
#include <hip/hip_runtime.h>
#include <math.h>

#define BATCH   4096
#define HIDDEN  64
#define BN_EPS  1e-5f
#define PI_F    3.14159265358979323846f

typedef __attribute__((ext_vector_type(2))) float v2f;
typedef __attribute__((ext_vector_type(8))) float v8f;

// ---------------- 4-qubit register-resident state-vector gates ----------------
// idx bit position of wire w is (3-w); all indices compile-time constant so the
// 16 complex amplitudes stay in VGPRs.

template<int BP>
__device__ __forceinline__ void rx_gate(float (&re)[16], float (&im)[16], float c, float s) {
#pragma unroll
  for (int i = 0; i < 16; ++i) {
    if ((i & (1 << BP)) == 0) {
      const int j = i | (1 << BP);
      float r0 = re[i], i0 = im[i], r1 = re[j], i1 = im[j];
      re[i] = fmaf(c, r0,  s * i1);   // c*a0 - i*s*a1
      im[i] = fmaf(c, i0, -s * r1);
      re[j] = fmaf(c, r1,  s * i0);   // -i*s*a0 + c*a1
      im[j] = fmaf(c, i1, -s * r0);
    }
  }
}

template<int BP>
__device__ __forceinline__ void rz_gate(float (&re)[16], float (&im)[16], float c, float s) {
#pragma unroll
  for (int i = 0; i < 16; ++i) {
    if ((i & (1 << BP)) == 0) {
      const int j = i | (1 << BP);
      float r0 = re[i], i0 = im[i], r1 = re[j], i1 = im[j];
      re[i] = fmaf(c, r0,  s * i0);   // (c - i s) * a0
      im[i] = fmaf(c, i0, -s * r0);
      re[j] = fmaf(c, r1, -s * i1);   // (c + i s) * a1
      im[j] = fmaf(c, i1,  s * r1);
    }
  }
}

template<int BPC, int BPT>
__device__ __forceinline__ void cnot_gate(float (&re)[16], float (&im)[16]) {
#pragma unroll
  for (int i = 0; i < 16; ++i) {
    if ((i & (1 << BPC)) != 0 && (i & (1 << BPT)) == 0) {
      const int j = i | (1 << BPT);
      float tr = re[i], ti = im[i];
      re[i] = re[j]; im[i] = im[j];
      re[j] = tr;    im[j] = ti;
    }
  }
}

// fast hw sincos: v_sin_f32 / v_cos_f32 (args here are bounded, |x| <= pi/2)
__device__ __forceinline__ void fsincos(float x, float* s, float* c) {
  *s = __sinf(x);
  *c = __cosf(x);
}

// ---------------- kernel 1: quantum features ----------------
// grid 1024 x 256: blockIdx>>8 = filter (uniform per block -> scalar qweights),
// lane task n = b*16 + p.  Output feats[b*64 + f*16 + p].
__global__ void __launch_bounds__(256)
qfeat_kernel(const float* __restrict__ x, const float* __restrict__ qweights,
             float* __restrict__ feats) {
  const int f = blockIdx.x >> 8;
  const int n = ((blockIdx.x & 255) << 8) | threadIdx.x;
  const int b = n >> 4;
  const int p = n & 15;

  // initial state: RX(x*pi)|0> per qubit -> outer product
  float ch[4], sh[4];
#pragma unroll
  for (int q = 0; q < 4; ++q) {
    float half = x[b * 64 + q * 16 + p] * (PI_F * 0.5f);
    fsincos(half, &sh[q], &ch[q]);
  }

  float re[16], im[16];
#pragma unroll
  for (int i = 0; i < 16; ++i) {
    float r = 1.f, m = 0.f;
#pragma unroll
    for (int q = 0; q < 4; ++q) {
      int bit = (i >> (3 - q)) & 1;
      float fr = bit ? 0.f : ch[q];
      float fi = bit ? -sh[q] : 0.f;
      float nr = r * fr - m * fi;
      float nm = r * fi + m * fr;
      r = nr; m = nm;
    }
    re[i] = r; im[i] = m;
  }

  const float* qw = qweights + f * 32;
#pragma unroll
  for (int layer = 0; layer < 4; ++layer) {
    const float* w = qw + layer * 8;
    float c, s;
    fsincos(w[0] * 0.5f, &s, &c); rx_gate<3>(re, im, c, s);
    fsincos(w[1] * 0.5f, &s, &c); rx_gate<2>(re, im, c, s);
    fsincos(w[2] * 0.5f, &s, &c); rx_gate<1>(re, im, c, s);
    fsincos(w[3] * 0.5f, &s, &c); rx_gate<0>(re, im, c, s);
    fsincos(w[4] * 0.5f, &s, &c); rz_gate<3>(re, im, c, s);
    fsincos(w[5] * 0.5f, &s, &c); rz_gate<2>(re, im, c, s);
    fsincos(w[6] * 0.5f, &s, &c); rz_gate<1>(re, im, c, s);
    fsincos(w[7] * 0.5f, &s, &c); rz_gate<0>(re, im, c, s);
    // CNOT wire pairs (3,2),(2,1),(1,0) -> bit pairs (0,1),(1,2),(2,3)
    cnot_gate<0, 1>(re, im);
    cnot_gate<1, 2>(re, im);
    cnot_gate<2, 3>(re, im);
  }

  // <Z_0>: wire 0 is bit 3
  float z = 0.f;
#pragma unroll
  for (int i = 0; i < 16; ++i) {
    float pr = fmaf(re[i], re[i], im[i] * im[i]);
    z += (i & 8) ? -pr : pr;
  }
  feats[b * 64 + f * 16 + p] = z;
}

// ---------------- kernel 2: fc1 GEMM via fp32 WMMA + bias + ReLU + BN partials ----
// H[i][c] = relu(sum_k F[i][k] * W[c][k] + bias[c]).
// One 16x16 tile per wave; K=64 -> 16x V_WMMA_F32_16X16X4_F32.
__global__ void __launch_bounds__(256)
fc1_wmma_kernel(const float* __restrict__ feats, const float* __restrict__ w,
                const float* __restrict__ bias, float* __restrict__ h,
                float* __restrict__ colsum, float* __restrict__ colsq) {
  const int lane = threadIdx.x & 31;
  const int wave = threadIdx.x >> 5;
  const int tile = blockIdx.x * 8 + wave;   // 1024 tiles total
  const int tm = tile >> 2;                 // 0..255 (batch tiles)
  const int tn = tile & 3;                  // 0..3   (hidden tiles)
  const int half = lane >> 4;
  const int l16  = lane & 15;

  const int arow = tm * 16 + l16;           // A: lanes 0-15 hold K=0,1; 16-31 hold K=2,3
  const int wrow = tn * 16 + l16;           // B[k][n] = W[n][k], same K split per lane half

  v8f acc = {};
#pragma unroll
  for (int ko = 0; ko < 64; ko += 4) {
    const int k = ko + half * 2;
    v2f a  = *(const v2f*)(feats + arow * 64 + k);
    v2f bb = *(const v2f*)(w     + wrow * 64 + k);
    acc = __builtin_amdgcn_wmma_f32_16x16x4_f32(
        /*neg_a=*/false, a, /*neg_b=*/false, bb,
        /*c_mod=*/(short)0, acc, /*reuse_a=*/false, /*reuse_b=*/false);
  }

  // D layout: VGPR v -> M = v + 8*half, N = l16
  const int col = tn * 16 + l16;
  const float bcol = bias[col];
  float s = 0.f, sq = 0.f;
#pragma unroll
  for (int v = 0; v < 8; ++v) {
    const int row = tm * 16 + v + half * 8;
    float val = acc[v] + bcol;
    val = val > 0.f ? val : 0.f;
    h[row * 64 + col] = val;
    s  += val;
    sq += val * val;
  }
  atomicAdd(colsum + col, s);
  atomicAdd(colsq  + col, sq);
}

// ---------------- kernel 0: zero BN accumulators ----------------
__global__ void zero_kernel(float* __restrict__ p, int n) {
  int i = blockIdx.x * blockDim.x + threadIdx.x;
  if (i < n) p[i] = 0.f;
}

// ---------------- kernel 3: fold BN + fc2 into 64 coeffs + constant ----------
__global__ void __launch_bounds__(64)
coef_kernel(const float* __restrict__ colsum, const float* __restrict__ colsq,
            const float* __restrict__ gamma, const float* __restrict__ beta,
            const float* __restrict__ fc2_w, const float* __restrict__ fc2_b,
            float* __restrict__ coef) {
  __shared__ float sh[64];
  const int j = threadIdx.x;
  const float inv_b = 1.f / (float)BATCH;
  float mean = colsum[j] * inv_b;
  float var  = colsq[j] * inv_b - mean * mean;
  float inv  = rsqrtf(var + BN_EPS);
  float scale = gamma[j] * inv;
  float wj = fc2_w[j];
  coef[j] = scale * wj;                       // multiplies h
  sh[j] = (beta[j] - mean * scale) * wj;      // goes into the constant
  __syncthreads();
  if (j == 0) {
    float c = fc2_b[0];
    for (int i = 0; i < 64; ++i) c += sh[i];
    coef[64] = c;
  }
}

// ---------------- kernel 4: logits = h . a + c ----------------
__global__ void __launch_bounds__(256)
fc2_kernel(const float* __restrict__ h, const float* __restrict__ coef,
           float* __restrict__ out) {
  __shared__ float a[72];
  if (threadIdx.x < 65) a[threadIdx.x] = coef[threadIdx.x];
  __syncthreads();
  const int row = blockIdx.x * blockDim.x + threadIdx.x;
  const float4* hp = (const float4*)(h + row * 64);
  float acc = a[64];
#pragma unroll
  for (int q = 0; q < 16; ++q) {
    float4 v = hp[q];
    acc = fmaf(v.x, a[q * 4 + 0], acc);
    acc = fmaf(v.y, a[q * 4 + 1], acc);
    acc = fmaf(v.z, a[q * 4 + 2], acc);
    acc = fmaf(v.w, a[q * 4 + 3], acc);
  }
  out[row] = acc;
}

// ---------------- launch ----------------
extern "C" void kernel_launch(void* const* d_in, const int* in_sizes, int n_in,
                              void* d_out, int out_size, void* d_ws, size_t ws_size,
                              hipStream_t stream) {
  const float* x      = (const float*)d_in[0];
  const float* qw     = (const float*)d_in[1];
  const float* fc1_w  = (const float*)d_in[2];
  const float* fc1_b  = (const float*)d_in[3];
  const float* gamma  = (const float*)d_in[4];
  const float* beta   = (const float*)d_in[5];
  const float* fc2_w  = (const float*)d_in[6];
  const float* fc2_b  = (const float*)d_in[7];
  float* out = (float*)d_out;

  float* ws     = (float*)d_ws;
  float* feats  = ws;                 // 4096*64
  float* h      = ws + 262144;        // 4096*64
  float* colsum = ws + 524288;        // 64
  float* colsq  = ws + 524352;        // 64
  float* coef   = ws + 524416;        // 65

  zero_kernel<<<1, 128, 0, stream>>>(colsum, 128);            // colsum + colsq
  qfeat_kernel<<<1024, 256, 0, stream>>>(x, qw, feats);
  fc1_wmma_kernel<<<128, 256, 0, stream>>>(feats, fc1_w, fc1_b, h, colsum, colsq);
  coef_kernel<<<1, 64, 0, stream>>>(colsum, colsq, gamma, beta, fc2_w, fc2_b, coef);
  fc2_kernel<<<16, 256, 0, stream>>>(h, coef, out);
}